// GCNModelVAE_67774583931169
// MI455X (gfx1250) — compile-verified
//
#include <hip/hip_runtime.h>
#include <hip/hip_bf16.h>

typedef float v2f __attribute__((ext_vector_type(2)));
typedef float v8f __attribute__((ext_vector_type(8)));

#define NN 10000
#define EE 320000
#define FF 512
#define HH1 32
#define HH2 16

// ---------------- zero the atomic-accumulation buffers ----------------
__global__ void vae_zero_kernel(float* __restrict__ h1acc,
                                float* __restrict__ zmean,
                                float* __restrict__ zlstd) {
    int i = blockIdx.x * blockDim.x + threadIdx.x;
    if (i < NN * HH1) h1acc[i] = 0.0f;
    if (i < NN * HH2) { zmean[i] = 0.0f; zlstd[i] = 0.0f; }
}

// ---------------- GEMM1: XW0 = features @ W0  [N,512]x[512,32] --------
// one wave per 16x16 output tile, K-loop of 128 x V_WMMA_F32_16X16X4_F32
__global__ void vae_gemm_xw0_kernel(const float* __restrict__ X,
                                    const float* __restrict__ W0,
                                    float* __restrict__ out) {
    const int wave = (blockIdx.x * blockDim.x + threadIdx.x) >> 5;
    const int ntiles = (NN / 16) * (HH1 / 16);   // 625 * 2
    if (wave >= ntiles) return;                  // uniform per wave: EXEC stays all-1
    const int tm = wave >> 1;
    const int tn = wave & 1;
    const int m0 = tm * 16, n0 = tn * 16;
    const int lane = threadIdx.x & 31;
    const int mr = lane & 15;     // A row / B col within tile
    const int kh = lane >> 4;     // K half-select

    v8f c = {};
    for (int k = 0; k < FF; k += 4) {
        const int ka = k + 2 * kh;
        v2f a = *(const v2f*)(X + (m0 + mr) * FF + ka);       // global_load_b64
        v2f b;
        b.x = W0[ka * HH1 + n0 + mr];
        b.y = W0[(ka + 1) * HH1 + n0 + mr];
        c = __builtin_amdgcn_wmma_f32_16x16x4_f32(false, a, false, b,
                                                  (short)0, c, false, false);
    }
    const int row0 = m0 + 8 * kh;
    const int col  = n0 + mr;
#pragma unroll
    for (int j = 0; j < 8; ++j)
        out[(row0 + j) * HH1 + col] = c[j];
}

// ---------------- SpMM #1: h1acc[dst] += w * xw0[src]  (atomics) ------
__global__ void vae_spmm1_kernel(const int* __restrict__ ei,
                                 const float* __restrict__ ew,
                                 const float* __restrict__ xw0,
                                 float* __restrict__ h1acc) {
    int idx = blockIdx.x * blockDim.x + threadIdx.x;
    if (idx >= EE * HH1) return;
    int e = idx >> 5;           // /32
    int f = idx & 31;
    int src = ei[e];
    int dst = ei[EE + e];
    atomicAdd(&h1acc[dst * HH1 + f], ew[e] * xw0[src * HH1 + f]);
}

// ---------------- ReLU in place ---------------------------------------
__global__ void vae_relu_kernel(float* __restrict__ h1) {
    int i = blockIdx.x * blockDim.x + threadIdx.x;
    if (i < NN * HH1) h1[i] = fmaxf(h1[i], 0.0f);
}

// ---------------- GEMM2: hm = h1@W_mean, hs = h1@W_std (K=32,N=16) ----
// one wave per 16-row tile; two accumulators share the A loads
__global__ void vae_gemm_heads_kernel(const float* __restrict__ h1,
                                      const float* __restrict__ Wm,
                                      const float* __restrict__ Ws,
                                      float* __restrict__ hm,
                                      float* __restrict__ hs) {
    const int wave = (blockIdx.x * blockDim.x + threadIdx.x) >> 5;
    if (wave >= NN / 16) return;
    const int m0 = wave * 16;
    const int lane = threadIdx.x & 31;
    const int mr = lane & 15;
    const int kh = lane >> 4;

    v8f cm = {}, cs = {};
    for (int k = 0; k < HH1; k += 4) {
        const int ka = k + 2 * kh;
        v2f a = *(const v2f*)(h1 + (m0 + mr) * HH1 + ka);
        v2f bm, bs;
        bm.x = Wm[ka * HH2 + mr];       bm.y = Wm[(ka + 1) * HH2 + mr];
        bs.x = Ws[ka * HH2 + mr];       bs.y = Ws[(ka + 1) * HH2 + mr];
        cm = __builtin_amdgcn_wmma_f32_16x16x4_f32(false, a, false, bm,
                                                   (short)0, cm, false, false);
        cs = __builtin_amdgcn_wmma_f32_16x16x4_f32(false, a, false, bs,
                                                   (short)0, cs, false, false);
    }
    const int row0 = m0 + 8 * kh;
#pragma unroll
    for (int j = 0; j < 8; ++j) {
        hm[(row0 + j) * HH2 + mr] = cm[j];
        hs[(row0 + j) * HH2 + mr] = cs[j];
    }
}

// ---------------- SpMM #2 (both heads) --------------------------------
__global__ void vae_spmm2_kernel(const int* __restrict__ ei,
                                 const float* __restrict__ ew,
                                 const float* __restrict__ hm,
                                 const float* __restrict__ hs,
                                 float* __restrict__ zmean,
                                 float* __restrict__ zlstd) {
    int idx = blockIdx.x * blockDim.x + threadIdx.x;
    if (idx >= EE * HH2) return;
    int e = idx >> 4;           // /16
    int j = idx & 15;
    int src = ei[e];
    int dst = ei[EE + e];
    float w = ew[e];
    atomicAdd(&zmean[dst * HH2 + j], w * hm[src * HH2 + j]);
    atomicAdd(&zlstd[dst * HH2 + j], w * hs[src * HH2 + j]);
}

// ---------------- reparameterization: z = mu + eps*exp(logstd) --------
__global__ void vae_reparam_kernel(const float* __restrict__ zmean,
                                   const float* __restrict__ zlstd,
                                   const float* __restrict__ eps,
                                   float* __restrict__ z) {
    int i = blockIdx.x * blockDim.x + threadIdx.x;
    if (i < NN * HH2)
        z[i] = zmean[i] + eps[i] * __expf(zlstd[i]);
}

// ---------------- decoder: recon = z @ z.T  [10000,10000], K=16 -------
// store-bandwidth bound (400 MB out) -> non-temporal stores; z (640 KB)
// stays hot in L2 for the 390K tile reads.
__global__ void vae_recon_kernel(const float* __restrict__ z,
                                 float* __restrict__ recon) {
    const int wave = (blockIdx.x * blockDim.x + threadIdx.x) >> 5;
    const int mt = NN / 16;                      // 625
    if (wave >= mt * mt) return;
    const int tm = wave / mt;
    const int tn = wave - tm * mt;
    const int m0 = tm * 16, n0 = tn * 16;
    const int lane = threadIdx.x & 31;
    const int mr = lane & 15;
    const int kh = lane >> 4;

    v8f c = {};
#pragma unroll
    for (int k = 0; k < HH2; k += 4) {
        const int ka = k + 2 * kh;
        v2f a = *(const v2f*)(z + (m0 + mr) * HH2 + ka);   // A tile row
        v2f b = *(const v2f*)(z + (n0 + mr) * HH2 + ka);   // B = z^T -> contiguous too
        c = __builtin_amdgcn_wmma_f32_16x16x4_f32(false, a, false, b,
                                                  (short)0, c, false, false);
    }
    const int row0 = m0 + 8 * kh;
    const int col  = n0 + mr;
#pragma unroll
    for (int j = 0; j < 8; ++j)
        __builtin_nontemporal_store(c[j], &recon[(size_t)(row0 + j) * NN + col]);
}

// ======================================================================
extern "C" void kernel_launch(void* const* d_in, const int* in_sizes, int n_in,
                              void* d_out, int out_size, void* d_ws, size_t ws_size,
                              hipStream_t stream) {
    const float* features = (const float*)d_in[0];   // [N, F]
    const int*   ei       = (const int*)d_in[1];     // [2, E]
    const float* ew       = (const float*)d_in[2];   // [E]
    const float* eps      = (const float*)d_in[3];   // [N, H2]
    const float* W0       = (const float*)d_in[4];   // [F, H1]
    const float* Wm       = (const float*)d_in[5];   // [H1, H2]
    const float* Ws       = (const float*)d_in[6];   // [H1, H2]

    float* recon = (float*)d_out;                            // [N*N]
    float* zmean = recon + (size_t)NN * NN;                  // [N, H2]
    float* zlstd = zmean + (size_t)NN * HH2;                 // [N, H2]

    float* ws   = (float*)d_ws;
    float* xw0   = ws;                          // [N, H1]
    float* h1acc = xw0   + NN * HH1;            // [N, H1]
    float* hm    = h1acc + NN * HH1;            // [N, H2]
    float* hs    = hm    + NN * HH2;            // [N, H2]
    float* z     = hs    + NN * HH2;            // [N, H2]

    const int B = 256;

    // 0) zero atomic accumulators (every call: graph replays re-use buffers)
    vae_zero_kernel<<<(NN * HH1 + B - 1) / B, B, 0, stream>>>(h1acc, zmean, zlstd);

    // 1) XW0 = features @ W0   (1250 wave-tiles)
    {
        int tiles = (NN / 16) * (HH1 / 16);
        int waves_per_block = B / 32;
        vae_gemm_xw0_kernel<<<(tiles + waves_per_block - 1) / waves_per_block, B, 0, stream>>>(
            features, W0, xw0);
    }

    // 2) h1 = spmm(adj, XW0) via f32 atomics, then ReLU
    vae_spmm1_kernel<<<(EE * HH1 + B - 1) / B, B, 0, stream>>>(ei, ew, xw0, h1acc);
    vae_relu_kernel<<<(NN * HH1 + B - 1) / B, B, 0, stream>>>(h1acc);

    // 3) hm = h1@W_mean, hs = h1@W_std  (625 wave-tiles, shared A loads)
    {
        int tiles = NN / 16;
        int waves_per_block = B / 32;
        vae_gemm_heads_kernel<<<(tiles + waves_per_block - 1) / waves_per_block, B, 0, stream>>>(
            h1acc, Wm, Ws, hm, hs);
    }

    // 4) z_mean / z_log_std = spmm(adj, hm/hs)
    vae_spmm2_kernel<<<(EE * HH2 + B - 1) / B, B, 0, stream>>>(ei, ew, hm, hs, zmean, zlstd);

    // 5) z = z_mean + eps * exp(z_log_std)
    vae_reparam_kernel<<<(NN * HH2 + B - 1) / B, B, 0, stream>>>(zmean, zlstd, eps, z);

    // 6) recon = z @ z.T  (390625 wave-tiles; NT stores for the 400 MB stream)
    {
        int tiles = (NN / 16) * (NN / 16);
        int waves_per_block = B / 32;
        vae_recon_kernel<<<(tiles + waves_per_block - 1) / waves_per_block, B, 0, stream>>>(
            z, recon);
    }
}